// ICON_Transformer_69810398429234
// MI455X (gfx1250) — compile-verified
//
#include <hip/hip_runtime.h>

typedef float v2f __attribute__((ext_vector_type(2)));
typedef float v8f __attribute__((ext_vector_type(8)));

#define WAVES 8   // 8 waves per block = 256 threads; one wave handles one cell

namespace {
constexpr int   kNH     = 7;
constexpr int   kE      = 16;
constexpr float kKappa  = 4.0f;
constexpr float kInvSig = 100.0f;               // 1 / SIGMA_D
constexpr float kDStep  = 0.05f / 3.0f;         // linspace(0, 0.05, 4) step
constexpr float kPi     = 3.14159265358979323846f;
constexpr float kTStep  = kPi / 4.0f;           // linspace(-pi, pi, 9)[:-1] step
constexpr float kClip   = 1.0f - 1e-7f;
}

template <int BB>
__global__ __launch_bounds__(WAVES * 32) void icon_proj_wmma(
    const float* __restrict__ x,       // [BB, N, 16]
    const float* __restrict__ coords,  // [2, N]  (lon row, lat row)
    const int*   __restrict__ adjc,    // [N, 7]
    float*       __restrict__ out,     // [BB, N, 4, 8, 16]
    int N)
{
  // Per-wave normalized weight tile W[32 bins][K padded to 8], 1 KB/wave.
  __shared__ float ldsw[WAVES][32][8];

  const int lane = threadIdx.x & 31;
  const int wave = threadIdx.x >> 5;
  int n = blockIdx.x * WAVES + wave;
  if (n >= N) n = N - 1;            // duplicate tail work; keeps waves at the barrier
  n = __builtin_amdgcn_readfirstlane(n);  // wave-uniform -> SGPR addressing

  // ---- lane k (k < 7) computes great-circle (d_k, phi_k) for neighbor k ----
  const int kq    = (lane < kNH) ? lane : (kNH - 1);
  const int idx_c = adjc[n * kNH];          // center (neighbor 0)
  const int idx_k = adjc[n * kNH + kq];
  const float lon1 = coords[idx_c], lat1 = coords[N + idx_c];
  const float lon2 = coords[idx_k], lat2 = coords[N + idx_k];
  float s1, c1, s2, c2, sdl, cdl;
  __sincosf(lat1, &s1, &c1);
  __sincosf(lat2, &s2, &c2);
  __sincosf(lon2 - lon1, &sdl, &cdl);
  float cosd = s1 * s2 + c1 * c2 * cdl;
  cosd = fminf(fmaxf(cosd, -kClip), kClip);
  const float d   = acosf(cosd);
  const float phi = atan2f(sdl * c2, c1 * s2 - s1 * c2 * cdl);

  // ---- broadcast neighbor indices + geometry across the wave ----
  // nbr[k] is identical in every lane -> pin to SGPRs for scalar addressing.
  int   nbr[kNH];
  float dk[kNH], pk[kNH];
#pragma unroll
  for (int k = 0; k < kNH; ++k) {
    nbr[k] = __builtin_amdgcn_readfirstlane(__shfl(idx_k, k, 32));
    dk[k]  = __shfl(d,   k, 32);
    pk[k]  = __shfl(phi, k, 32);
  }

  // ---- per-lane weights: lane = flattened (dist, theta) bin = GEMM M-row ----
  const float dist  = (float)(lane >> 3) * kDStep;
  const float theta = -kPi + (float)(lane & 7) * kTStep;
  float w[kNH];
  float wsum = 0.0f;
#pragma unroll
  for (int k = 0; k < kNH; ++k) {
    const float dd = (dist - dk[k]) * kInvSig;
    // vm * nd == exp(kappa*cos(theta-phi) - 0.5*((dist-d)/sigma)^2)
    w[k] = __expf(kKappa * __cosf(theta - pk[k]) - 0.5f * dd * dd);
    wsum += w[k];
  }
  const float inv = 1.0f / (wsum + 1e-10f);
#pragma unroll
  for (int k = 0; k < kNH; ++k) ldsw[wave][lane][k] = w[k] * inv;
  ldsw[wave][lane][7] = 0.0f;  // zero K-pad: makes the 8th B-row a don't-care
  __syncthreads();

  const int col = lane & 15;   // WMMA N index == feature e
  const int hi  = lane >> 4;   // lane half

  // ==== C[32x16] = W[32x8] * X[8x16], exact f32, V_WMMA_F32_16X16X4 ====
  // A 16x4 f32 layout: VGPR0 = K {0|2} (lo|hi lane half), VGPR1 = K {1|3}.
  v2f afrag[2][2];  // [m-tile][k-step]
#pragma unroll
  for (int mt = 0; mt < 2; ++mt)
#pragma unroll
    for (int kt = 0; kt < 2; ++kt) {
      const int row = 16 * mt + col;
      const int kb  = 4 * kt + 2 * hi;
      afrag[mt][kt].x = ldsw[wave][row][kb];
      afrag[mt][kt].y = ldsw[wave][row][kb + 1];
    }

#pragma unroll
  for (int b = 0; b < BB; ++b) {
    const float* xb = x + (size_t)b * N * kE;
    // B 4x16 f32 fragment (mirror of A layout, lane = column).
    // K=7 row is never weighted (A pad is 0) -> clamp index, load unconditionally.
    v2f bfrag[2];
#pragma unroll
    for (int kt = 0; kt < 2; ++kt) {
      const int k0 = 4 * kt + 2 * hi;                    // 0,2 / 4,6
      const int k1 = (k0 + 1 < kNH) ? (k0 + 1) : (kNH - 1);
      bfrag[kt].x = xb[(unsigned)nbr[k0] * kE + col];
      bfrag[kt].y = xb[(unsigned)nbr[k1] * kE + col];
    }

    v8f acc0 = {};
    v8f acc1 = {};
    acc0 = __builtin_amdgcn_wmma_f32_16x16x4_f32(false, afrag[0][0], false, bfrag[0], (short)0, acc0, false, false);
    acc0 = __builtin_amdgcn_wmma_f32_16x16x4_f32(false, afrag[0][1], false, bfrag[1], (short)0, acc0, false, false);
    acc1 = __builtin_amdgcn_wmma_f32_16x16x4_f32(false, afrag[1][0], false, bfrag[0], (short)0, acc1, false, false);
    acc1 = __builtin_amdgcn_wmma_f32_16x16x4_f32(false, afrag[1][1], false, bfrag[1], (short)0, acc1, false, false);

    // C layout: VGPR r -> M = r + 8*hi, N = col; out[b,n,bin,e], bin = M (+16 tile 1).
    float* op = out + ((size_t)b * N + n) * 512 + (unsigned)(hi * 128 + col);
#pragma unroll
    for (int r = 0; r < 8; ++r) {
      op[r * 16]       = acc0[r];
      op[256 + r * 16] = acc1[r];
    }
  }
}

extern "C" void kernel_launch(void* const* d_in, const int* in_sizes, int n_in,
                              void* d_out, int out_size, void* d_ws, size_t ws_size,
                              hipStream_t stream) {
  const float* x      = (const float*)d_in[0];  // [B, N, 16] f32
  const float* coords = (const float*)d_in[1];  // [2, N]     f32
  const int*   adjc   = (const int*)d_in[2];    // [N, 7]     i32
  float*       out    = (float*)d_out;          // [B, N, 4, 8, 16] f32

  const int N = in_sizes[2] / kNH;
  const int B = in_sizes[0] / (N * kE);

  const int blocks = (N + WAVES - 1) / WAVES;   // one wave per cell

  if (B == 2) {
    icon_proj_wmma<2><<<blocks, WAVES * 32, 0, stream>>>(x, coords, adjc, out, N);
  } else {
    // generic fallback: one single-batch pass per batch slice
    for (int b = 0; b < B; ++b) {
      icon_proj_wmma<1><<<blocks, WAVES * 32, 0, stream>>>(
          x + (size_t)b * N * kE, coords, adjc, out + (size_t)b * N * 512, N);
    }
  }
}